// WeightedDiceLoss_46729244180574
// MI455X (gfx1250) — compile-verified
//
#include <hip/hip_runtime.h>
#include <stdint.h>

#define WAVES  8
#define BLOCK  256
#define GRID   2048
#define NBUF   4          // async ring depth (3 stages in flight + 1 being consumed)

// ---------------------------------------------------------------------------
// CDNA5 async global->LDS staging (ASYNCcnt-tracked). Per lane: 16 bytes.
//   LDS address from the vdst VGPR; global address = SGPR base + VGPR byte offset
// Read-once stream -> non-temporal hint (805 MB >> 192 MB L2).
// ---------------------------------------------------------------------------
__device__ __forceinline__ void async_b128(const void* sbase, unsigned int voff_bytes,
                                           void* lds_ptr) {
    unsigned int laddr = (unsigned int)(uintptr_t)lds_ptr; // low 32 bits = LDS offset
    asm volatile("global_load_async_to_lds_b128 %0, %1, %2 th:TH_LOAD_NT"
                 :: "v"(laddr), "v"(voff_bytes), "s"(sbase)
                 : "memory");
}
// Async loads complete ("done") in order -> waiting for <= 3*k leaves the k
// newest 3-instruction stages outstanding and guarantees older stages landed.
__device__ __forceinline__ void wait_async_le9() { asm volatile("s_wait_asynccnt 0x9" ::: "memory"); }
__device__ __forceinline__ void wait_async_le6() { asm volatile("s_wait_asynccnt 0x6" ::: "memory"); }
__device__ __forceinline__ void wait_async_le3() { asm volatile("s_wait_asynccnt 0x3" ::: "memory"); }
__device__ __forceinline__ void wait_async_le0() { asm volatile("s_wait_asynccnt 0x0" ::: "memory"); }

// Bank-conflict-free per-wave LDS binning:
// slot(b, lane) = (b>>1)*64 + 2*lane + (b&1)  -> bank = 2*lane (+0/1), exclusive per lane.
// counts live at slot+192 words (same bank pair, different instruction -> no conflict).
__device__ __forceinline__ void bin1(float* mybins, int lane2, float p, int t, int v) {
    int b    = t * 3 + v;                       // 0..5
    int slot = ((b >> 1) << 6) + lane2 + (b & 1);
    atomicAdd(&mybins[slot],       p);          // ds_add_f32 (sum)
    atomicAdd(&mybins[slot + 192], 1.0f);       // ds_add_f32 (count)
}

__global__ __launch_bounds__(BLOCK) void wdl_zero(float* ws) {
    if (threadIdx.x < 12) ws[threadIdx.x] = 0.0f;
}

__global__ __launch_bounds__(BLOCK) void wdl_main(const float* __restrict__ pred,
                                                  const int*   __restrict__ tgt,
                                                  const int*   __restrict__ wgt,
                                                  float*       __restrict__ ws,
                                                  unsigned long long nElem) {
    __shared__ float4 sp[WAVES][NBUF][32];
    __shared__ int4   st[WAVES][NBUF][32];
    __shared__ int4   sw[WAVES][NBUF][32];
    __shared__ float  bins[WAVES][384];   // [0,192): sums  [192,384): counts

    const int tid   = threadIdx.x;
    const int w     = tid >> 5;
    const int lane  = tid & 31;
    const int lane2 = lane << 1;

    // zero this block's bins
    float* bflat = &bins[0][0];
    for (int k = tid; k < WAVES * 384; k += BLOCK) bflat[k] = 0.0f;
    __syncthreads();

    const unsigned int nThreads = gridDim.x * BLOCK;
    const unsigned int gtid     = blockIdx.x * BLOCK + tid;
    const unsigned long long ngroups = nElem >> 2;                  // float4 groups
    const unsigned int iters = (unsigned int)(ngroups / nThreads);
    const unsigned int rem   = (unsigned int)(ngroups % nThreads);

    float* mybins = &bins[w][0];

    const unsigned int stride = nThreads * 16u;    // bytes per pipelined iteration
    unsigned int off          = gtid * 16u;        // byte offset of this lane's group

    // -------- 4-deep async ring, 3 stages (9 requests) in flight per wave ----
    {
        unsigned int poff = off;
        for (unsigned int s = 0; s < 3u && s < iters; ++s) {
            async_b128(pred, poff, &sp[w][s][lane]);
            async_b128(tgt,  poff, &st[w][s][lane]);
            async_b128(wgt,  poff, &sw[w][s][lane]);
            poff += stride;
        }
    }
    for (unsigned int i = 0; i < iters; ++i) {
        const int cur = (int)(i & (NBUF - 1u));
        const unsigned int ahead = iters - 1u - i;       // stages newer than i
        if (ahead >= 3u) {
            const unsigned int noff = off + 3u * stride;
            const int nb = (int)((i + 3u) & (NBUF - 1u));
            async_b128(pred, noff, &sp[w][nb][lane]);
            async_b128(tgt,  noff, &st[w][nb][lane]);
            async_b128(wgt,  noff, &sw[w][nb][lane]);
            wait_async_le9();
        } else if (ahead == 2u) {
            wait_async_le6();
        } else if (ahead == 1u) {
            wait_async_le3();
        } else {
            wait_async_le0();
        }
        const float4 p4 = sp[w][cur][lane];        // ds_load_b128
        const int4   t4 = st[w][cur][lane];
        const int4   v4 = sw[w][cur][lane];
        bin1(mybins, lane2, p4.x, t4.x, v4.x);
        bin1(mybins, lane2, p4.y, t4.y, v4.y);
        bin1(mybins, lane2, p4.z, t4.z, v4.z);
        bin1(mybins, lane2, p4.w, t4.w, v4.w);
        off += stride;
    }

    // -------- generic remainder: leftover float4 groups (direct loads) --------
    if (gtid < rem) {
        const unsigned long long g = (unsigned long long)iters * nThreads + gtid;
        const float4 p4 = ((const float4*)pred)[g];
        const int4   t4 = ((const int4*)tgt)[g];
        const int4   v4 = ((const int4*)wgt)[g];
        bin1(mybins, lane2, p4.x, t4.x, v4.x);
        bin1(mybins, lane2, p4.y, t4.y, v4.y);
        bin1(mybins, lane2, p4.z, t4.z, v4.z);
        bin1(mybins, lane2, p4.w, t4.w, v4.w);
    }
    // -------- scalar tail (nElem % 4): straight to global ws --------
    if (gtid == 0) {
        for (unsigned long long e = ngroups << 2; e < nElem; ++e) {
            const int b = tgt[e] * 3 + wgt[e];
            atomicAdd(&ws[b],     pred[e]);
            atomicAdd(&ws[6 + b], 1.0f);
        }
    }

    __syncthreads();

    // -------- block reduction: 8 waves x 32 lane-slots -> 12 scalars --------
    if (tid < 12) {
        const int b    = tid % 6;
        const int base = (tid / 6) * 192 + ((b >> 1) << 6) + (b & 1);
        float acc = 0.0f;
        for (int ww = 0; ww < WAVES; ++ww)
            for (int L = 0; L < 32; ++L)
                acc += bins[ww][base + (L << 1)];
        atomicAdd(&ws[tid], acc);                  // global_atomic_add_f32
    }
}

__global__ __launch_bounds__(BLOCK) void wdl_final(const float* __restrict__ ws,
                                                   float* __restrict__ out) {
    if (threadIdx.x == 0) {
        double num = 0.0, den = 0.0;
        for (int b = 0; b < 6; ++b) {
            const double s = (double)ws[b];
            const double n = (double)ws[6 + b];
            if (n > 0.0) {
                const double inv = 1.0 / (n * n);
                num += s * inv;
                den += (s + n) * inv;
            }
        }
        out[0] = (den > 0.0) ? (float)(1.0 - 2.0 * num / den) : 0.0f;
    }
}

extern "C" void kernel_launch(void* const* d_in, const int* in_sizes, int n_in,
                              void* d_out, int out_size, void* d_ws, size_t ws_size,
                              hipStream_t stream) {
    const float* pred = (const float*)d_in[0];
    const int*   tgt  = (const int*)d_in[1];
    const int*   wgt  = (const int*)d_in[2];
    float*       ws   = (float*)d_ws;
    const unsigned long long n = (unsigned long long)in_sizes[0];

    wdl_zero <<<1, BLOCK, 0, stream>>>(ws);
    wdl_main <<<GRID, BLOCK, 0, stream>>>(pred, tgt, wgt, ws, n);
    wdl_final<<<1, BLOCK, 0, stream>>>(ws, (float*)d_out);
}